// PredictorModel_87393994539239
// MI455X (gfx1250) — compile-verified
//
#include <hip/hip_runtime.h>
#include <hip/hip_bf16.h>
#include <math.h>

// ---- constants ----
#define NS   2048          // samples / FFT length
#define NT_  64            // transmit elements
#define NE_  32            // encodings
#define NR_  64            // receive elements
#define NB_  2             // batch
#define NP_  40000         // pixels
#define SIGS 8192          // NB_*NT_*NR_
#define TWO_PI 6.28318530717958647692f

typedef float v2f __attribute__((ext_vector_type(2)));
typedef float v8f __attribute__((ext_vector_type(8)));

__device__ __forceinline__ v8f wmma_f32(v2f a, v2f b, v8f c) {
    // V_WMMA_F32_16X16X4_F32: D = A(16x4) * B(4x16) + C(16x16)
    return __builtin_amdgcn_wmma_f32_16x16x4_f32(false, a, false, b, (short)0, c, false, false);
}

// CDNA5 async global->LDS copies (ASYNCcnt-tracked, no VGPR round-trip).
__device__ __forceinline__ void async_g2l_b32(void* lds, const void* gp) {
    unsigned l = (unsigned)(unsigned long long)lds;
    unsigned long long g = (unsigned long long)gp;
    asm volatile("global_load_async_to_lds_b32 %0, %1, off" :: "v"(l), "v"(g) : "memory");
}
__device__ __forceinline__ void async_g2l_b64(void* lds, const void* gp) {
    unsigned l = (unsigned)(unsigned long long)lds;
    unsigned long long g = (unsigned long long)gp;
    asm volatile("global_load_async_to_lds_b64 %0, %1, off" :: "v"(l), "v"(g) : "memory");
}
__device__ __forceinline__ void wait_async0() {
    asm volatile("s_wait_asynccnt 0" ::: "memory");
}

// ------------------------------------------------------------------
// Kernel 1: per-frequency H and Tikhonov pseudo-inverse (32x32 complex
// Gauss-Jordan with 64 RHS, all in LDS). One block per frequency.
// Stores HinvT[f][t][e] (row-major A-matrix for the decode WMMA).
// ------------------------------------------------------------------
__global__ void k_h_hinv(const float* __restrict__ delays, const float* __restrict__ weights,
                         float2* __restrict__ HinvT) {
    __shared__ float Hr[NT_*NE_], Hi_[NT_*NE_];   // H[t][e]
    __shared__ float Ar[NE_*NE_], Ai[NE_*NE_];    // A = H^H H + tik I
    __shared__ float Rr[NE_*NT_], Ri[NE_*NT_];    // RHS = H^H -> becomes Hinv
    __shared__ float facr[NE_], faci[NE_];
    int f = blockIdx.x;
    float fc = (f < NS/2 ? (float)f : (float)(f - NS)) / (float)NS;

    for (int idx = threadIdx.x; idx < NT_*NE_; idx += 256) {
        int t = idx >> 5, e = idx & 31;
        float d = delays[e*NT_ + t], w = weights[e*NT_ + t];
        float sn, cs; __sincosf(-TWO_PI * fc * d, &sn, &cs);
        Hr[idx] = w * cs; Hi_[idx] = w * sn;
    }
    __syncthreads();
    for (int idx = threadIdx.x; idx < NE_*NE_; idx += 256) {
        int i = idx >> 5, j = idx & 31;
        float sr = (i == j) ? 0.1f : 0.0f, si = 0.0f;
        for (int t = 0; t < NT_; ++t) {
            float hir = Hr[t*NE_+i], hii = Hi_[t*NE_+i];
            float hjr = Hr[t*NE_+j], hji = Hi_[t*NE_+j];
            sr += hir*hjr + hii*hji;   // conj(h_i) * h_j
            si += hir*hji - hii*hjr;
        }
        Ar[idx] = sr; Ai[idx] = si;
    }
    for (int idx = threadIdx.x; idx < NE_*NT_; idx += 256) {
        int i = idx >> 6, t = idx & 63;
        Rr[idx] = Hr[t*NE_+i]; Ri[idx] = -Hi_[t*NE_+i];   // H^H
    }
    __syncthreads();

    for (int k = 0; k < NE_; ++k) {
        float pr = Ar[k*NE_+k], pi = Ai[k*NE_+k];
        float ip = 1.0f / (pr*pr + pi*pi);
        float ivr = pr*ip, ivi = -pi*ip;
        for (int c = threadIdx.x; c < 96; c += 256) {
            if (c < 32) {
                float xr = Ar[k*NE_+c], xi = Ai[k*NE_+c];
                Ar[k*NE_+c] = xr*ivr - xi*ivi; Ai[k*NE_+c] = xr*ivi + xi*ivr;
            } else {
                int t = c - 32;
                float xr = Rr[k*NT_+t], xi = Ri[k*NT_+t];
                Rr[k*NT_+t] = xr*ivr - xi*ivi; Ri[k*NT_+t] = xr*ivi + xi*ivr;
            }
        }
        __syncthreads();
        if (threadIdx.x < NE_) { facr[threadIdx.x] = Ar[threadIdx.x*NE_+k];
                                 faci[threadIdx.x] = Ai[threadIdx.x*NE_+k]; }
        __syncthreads();
        for (int idx = threadIdx.x; idx < 32*96; idx += 256) {
            int i = idx / 96, c = idx - i*96;
            if (i == k) continue;
            float fr = facr[i], fi = faci[i];
            if (c < 32) {
                float xr = Ar[k*NE_+c], xi = Ai[k*NE_+c];
                Ar[i*NE_+c] -= fr*xr - fi*xi; Ai[i*NE_+c] -= fr*xi + fi*xr;
            } else {
                int t = c - 32;
                float xr = Rr[k*NT_+t], xi = Ri[k*NT_+t];
                Rr[i*NT_+t] -= fr*xr - fi*xi; Ri[i*NT_+t] -= fr*xi + fi*xr;
            }
        }
        __syncthreads();
    }
    for (int idx = threadIdx.x; idx < NT_*NE_; idx += 256) {
        int t = idx >> 5, e = idx & 31;
        HinvT[(size_t)f*(NT_*NE_) + idx] = make_float2(Rr[e*NT_+t], Ri[e*NT_+t]);
    }
}

// ------------------------------------------------------------------
// Forward FFT (length 2048, Stockham radix-2 in LDS), one signal/block.
// Strided input samples streamed straight into LDS with async copies.
// Output layout: F[f][sig], f-major so each per-frequency matmul reads a
// contiguous 64KB block.
// ------------------------------------------------------------------
__global__ void k_fft_fwd(const float* __restrict__ datas, float2* __restrict__ F) {
    __shared__ float xr[NS], xi[NS], yr_[NS], yi_[NS];
    int sig = blockIdx.x;
    int b = sig >> 12, t = (sig >> 6) & 63, r = sig & 63;
    const float* src = datas + (((size_t)(b*NT_ + t)) * NS) * NR_ + r;
    for (int s = threadIdx.x; s < NS; s += 256) {
        async_g2l_b32(&xr[s], &src[(size_t)s*NR_]);
        xi[s] = 0.0f;
    }
    wait_async0();
    __syncthreads();
    float *ar = xr, *ai = xi, *br = yr_, *bi = yi_;
    int n = NS, s = 1;
    for (int st = 0; st < 11; ++st) {
        int m = n >> 1;
        for (int j = threadIdx.x; j < NS/2; j += 256) {
            int p = j / s, q = j - p*s;
            float c, sn; __sincosf(-TWO_PI * (float)p / (float)n, &sn, &c);
            float a0r = ar[q + s*p],       a0i = ai[q + s*p];
            float a1r = ar[q + s*(p+m)],   a1i = ai[q + s*(p+m)];
            br[q + s*2*p]     = a0r + a1r;  bi[q + s*2*p]     = a0i + a1i;
            float dr = a0r - a1r, di = a0i - a1i;
            br[q + s*(2*p+1)] = dr*c - di*sn;  bi[q + s*(2*p+1)] = dr*sn + di*c;
        }
        __syncthreads();
        float* tp; tp = ar; ar = br; br = tp; tp = ai; ai = bi; bi = tp;
        n >>= 1; s <<= 1;
    }
    for (int f = threadIdx.x; f < NS; f += 256)
        F[(size_t)f*SIGS + sig] = make_float2(ar[f], ai[f]);
}

// ------------------------------------------------------------------
// Encode: Y[f] = H^T(f) * F(f)   (complex 32x64 * 64x64 per (f,b) block)
// B tile streamed global->LDS asynchronously while H is synthesized
// (sincos) into LDS; then V_WMMA_F32_16X16X4_F32, 4 per complex step.
// ------------------------------------------------------------------
__global__ void k_encode(const float* __restrict__ delays, const float* __restrict__ weights,
                         const float2* __restrict__ F, float2* __restrict__ Y) {
    __shared__ float  sAr[NE_*NT_], sAi[NE_*NT_];   // A[m=e][k=t]
    __shared__ float2 sB[NT_*NR_];                  // B[k=t][n=r] interleaved
    int f = blockIdx.x, b = blockIdx.y;
    float fc = (f < NS/2 ? (float)f : (float)(f - NS)) / (float)NS;

    // kick off async B-tile copy (contiguous 32KB), overlap with H sincos
    const float2* Fb = F + (size_t)f*SIGS + b*(NT_*NR_);
    for (int idx = threadIdx.x; idx < NT_*NR_; idx += 256)
        async_g2l_b64(&sB[idx], &Fb[idx]);

    for (int idx = threadIdx.x; idx < NE_*NT_; idx += 256) {
        float d = delays[idx], w = weights[idx];   // idx = e*64 + t
        float sn, cs; __sincosf(-TWO_PI * fc * d, &sn, &cs);
        sAr[idx] = w * cs; sAi[idx] = w * sn;
    }
    wait_async0();
    __syncthreads();

    int wv = threadIdx.x >> 5, lane = threadIdx.x & 31;
    int tileM = wv >> 2, tileN = wv & 3;
    int mrow = tileM*16 + (lane & 15);
    int ncol = tileN*16 + (lane & 15);
    int khalf = lane >> 4;
    v8f accR = {0,0,0,0,0,0,0,0}, accI = {0,0,0,0,0,0,0,0};
    for (int ks = 0; ks < 16; ++ks) {
        int ka = ks*4 + khalf*2;
        v2f arr_ = { sAr[mrow*NT_ + ka], sAr[mrow*NT_ + ka + 1] };
        v2f ari  = { sAi[mrow*NT_ + ka], sAi[mrow*NT_ + ka + 1] };
        int kb = ks*4 + khalf;
        float2 b0 = sB[kb*NR_ + ncol], b1 = sB[(kb+2)*NR_ + ncol];
        v2f brr  = { b0.x, b1.x };
        v2f bri  = { b0.y, b1.y };
        v2f brin = { -b0.y, -b1.y };
        accR = wmma_f32(arr_, brr,  accR);   // Re: Hr*Fr
        accR = wmma_f32(ari,  brin, accR);   //     - Hi*Fi
        accI = wmma_f32(arr_, bri,  accI);   // Im: Hr*Fi
        accI = wmma_f32(ari,  brr,  accI);   //     + Hi*Fr
    }
    float2* Yf = Y + (size_t)f*(NE_*2*NR_);
    for (int j = 0; j < 8; ++j) {
        int m = tileM*16 + khalf*8 + j;
        Yf[m*(2*NR_) + b*NR_ + ncol] = make_float2(accR[j], accI[j]);
    }
}

// ------------------------------------------------------------------
// Hermitian projection: F2[k] = 0.5*(Y[k] + conj(Y[N-k]))
// (fuses ifft -> real() -> fft). One thread owns the (k, N-k) pair.
// ------------------------------------------------------------------
__global__ void k_herm(float2* __restrict__ Y) {
    int idx = blockIdx.x*256 + threadIdx.x;
    if (idx >= (NS/2 + 1) * (NE_*2*NR_)) return;
    int k = idx >> 12, c = idx & 4095;
    int nk = (NS - k) & (NS - 1);
    size_t i1 = (size_t)k*4096 + c, i2 = (size_t)nk*4096 + c;
    float2 a = Y[i1], q = Y[i2];
    Y[i1] = make_float2(0.5f*(a.x + q.x), 0.5f*(a.y - q.y));
    Y[i2] = make_float2(0.5f*(q.x + a.x), 0.5f*(q.y - a.y));
}

// ------------------------------------------------------------------
// Decode: Z[f] = HinvT(f) * F2(f)  (complex 64x32 * 32x64 per (f,b) block)
// Both tiles streamed global->LDS asynchronously.
// ------------------------------------------------------------------
__global__ void k_decode(const float2* __restrict__ HinvT, const float2* __restrict__ Y,
                         float2* __restrict__ Z) {
    __shared__ float2 sA[NT_*NE_];   // A[m=t][k=e] interleaved
    __shared__ float2 sB[NE_*NR_];   // B[k=e][n=r] interleaved
    int f = blockIdx.x, b = blockIdx.y;
    const float2* Hf = HinvT + (size_t)f*(NT_*NE_);
    for (int idx = threadIdx.x; idx < NT_*NE_; idx += 256)
        async_g2l_b64(&sA[idx], &Hf[idx]);                   // idx = t*32 + e
    const float2* Yf = Y + (size_t)f*(NE_*2*NR_);
    for (int idx = threadIdx.x; idx < NE_*NR_; idx += 256) {
        int e = idx >> 6, r = idx & 63;
        async_g2l_b64(&sB[idx], &Yf[e*(2*NR_) + b*NR_ + r]);
    }
    wait_async0();
    __syncthreads();

    int wv = threadIdx.x >> 5, lane = threadIdx.x & 31;
    int khalf = lane >> 4;
    float2* Zf = Z + (size_t)f*SIGS;
    for (int it = 0; it < 2; ++it) {
        int tile = wv + it*8;
        int tileM = tile >> 2, tileN = tile & 3;
        int mrow = tileM*16 + (lane & 15);
        int ncol = tileN*16 + (lane & 15);
        v8f accR = {0,0,0,0,0,0,0,0}, accI = {0,0,0,0,0,0,0,0};
        for (int ks = 0; ks < 8; ++ks) {
            int ka = ks*4 + khalf*2;
            float2 a0 = sA[mrow*NE_ + ka], a1 = sA[mrow*NE_ + ka + 1];
            v2f arr_ = { a0.x, a1.x };
            v2f ari  = { a0.y, a1.y };
            int kb = ks*4 + khalf;
            float2 b0 = sB[kb*NR_ + ncol], b1 = sB[(kb+2)*NR_ + ncol];
            v2f brr  = { b0.x, b1.x };
            v2f bri  = { b0.y, b1.y };
            v2f brin = { -b0.y, -b1.y };
            accR = wmma_f32(arr_, brr,  accR);
            accR = wmma_f32(ari,  brin, accR);
            accI = wmma_f32(arr_, bri,  accI);
            accI = wmma_f32(ari,  brr,  accI);
        }
        for (int j = 0; j < 8; ++j) {
            int t = tileM*16 + khalf*8 + j;
            Zf[(b*NT_ + t)*NR_ + ncol] = make_float2(accR[j], accI[j]);
        }
    }
}

// ------------------------------------------------------------------
// Hermitian projection + Hilbert mask (fuses ifft->real->fft->*h):
// W[0]=Re(Z0); W[k]=Z[k]+conj(Z[N-k]) for 1<=k<N/2; W[N/2]=Re; W[k>N/2]=0.
// ------------------------------------------------------------------
__global__ void k_hermhilb(float2* __restrict__ Z) {
    int idx = blockIdx.x*256 + threadIdx.x;
    if (idx >= (NS/2 + 1) * SIGS) return;
    int k = idx >> 13, c = idx & 8191;
    int nk = (NS - k) & (NS - 1);
    size_t i1 = (size_t)k*SIGS + c, i2 = (size_t)nk*SIGS + c;
    float2 a = Z[i1], q = Z[i2];
    if (k == 0 || k == NS/2) {
        Z[i1] = make_float2(0.5f*(a.x + q.x), 0.5f*(a.y - q.y));
    } else {
        Z[i1] = make_float2(a.x + q.x, a.y - q.y);
        Z[i2] = make_float2(0.0f, 0.0f);
    }
}

// ------------------------------------------------------------------
// Inverse FFT -> analytic signal iq, stored [b][t][r][s] (S contiguous
// so the beamformer's i0/i0+1 gather hits adjacent float2s).
// Strided spectrum gathers streamed into LDS asynchronously.
// ------------------------------------------------------------------
__global__ void k_ifft(const float2* __restrict__ Z, float2* __restrict__ IQ) {
    __shared__ float2 sIn[NS];
    __shared__ float xr[NS], xi[NS], yr_[NS], yi_[NS];
    int sig = blockIdx.x;
    for (int s = threadIdx.x; s < NS; s += 256)
        async_g2l_b64(&sIn[s], &Z[(size_t)s*SIGS + sig]);
    wait_async0();
    __syncthreads();
    for (int s = threadIdx.x; s < NS; s += 256) {
        float2 v = sIn[s]; xr[s] = v.x; xi[s] = v.y;
    }
    __syncthreads();
    float *ar = xr, *ai = xi, *br = yr_, *bi = yi_;
    int n = NS, s = 1;
    for (int st = 0; st < 11; ++st) {
        int m = n >> 1;
        for (int j = threadIdx.x; j < NS/2; j += 256) {
            int p = j / s, q = j - p*s;
            float c, sn; __sincosf(TWO_PI * (float)p / (float)n, &sn, &c);
            float a0r = ar[q + s*p],     a0i = ai[q + s*p];
            float a1r = ar[q + s*(p+m)], a1i = ai[q + s*(p+m)];
            br[q + s*2*p]     = a0r + a1r;  bi[q + s*2*p]     = a0i + a1i;
            float dr = a0r - a1r, di = a0i - a1i;
            br[q + s*(2*p+1)] = dr*c - di*sn;  bi[q + s*(2*p+1)] = dr*sn + di*c;
        }
        __syncthreads();
        float* tp; tp = ar; ar = br; br = tp; tp = ai; ai = bi; bi = tp;
        n >>= 1; s <<= 1;
    }
    const float sc = 1.0f / (float)NS;
    float2* dst = IQ + (size_t)sig*NS;
    for (int ss = threadIdx.x; ss < NS; ss += 256)
        dst[ss] = make_float2(ar[ss]*sc, ai[ss]*sc);
}

// ------------------------------------------------------------------
// Delay-and-sum beamformer: one thread per (b, pixel); 64 element
// distances in registers; 64x64 fractional-delay complex gathers.
// ------------------------------------------------------------------
__global__ void k_beamform(const float2* __restrict__ IQ, const float* __restrict__ bf,
                           float* __restrict__ amp, unsigned int* __restrict__ mx) {
    int idx = blockIdx.x*256 + threadIdx.x;
    if (idx >= NB_*NP_) return;
    int b = idx / NP_, p = idx - b*NP_;
    float dd[64];
#pragma unroll
    for (int e = 0; e < 64; ++e) dd[e] = bf[e*NP_ + p];
    const float invDR = 20.0e6f / 1540.0f;   // fs / c
    float accr = 0.0f, acci = 0.0f;
    const float2* IQb = IQ + (size_t)b*NT_*NR_*NS;
    for (int t = 0; t < NT_; ++t) {
        float dt = dd[t];
        const float2* IQt = IQb + (size_t)t*NR_*NS;
#pragma unroll 4
        for (int r = 0; r < NR_; ++r) {
            float tf = (dt + dd[r]) * invDR;
            float fi = floorf(tf);
            int i0 = (int)fi;
            if (i0 >= 0 && i0 < NS-1) {
                float frac = tf - fi;
                const float2* sg = IQt + (size_t)r*NS;
                float2 v0 = sg[i0], v1 = sg[i0+1];
                accr += v0.x + (v1.x - v0.x)*frac;
                acci += v0.y + (v1.y - v0.y)*frac;
            }
        }
    }
    float a = sqrtf(accr*accr + acci*acci) + 1e-15f;
    amp[idx] = a;
    atomicMax(&mx[b], __float_as_uint(a));   // positive floats: bit order == value order
}

__global__ void k_finalize(const float* __restrict__ amp, const unsigned int* __restrict__ mx,
                           float* __restrict__ out) {
    int idx = blockIdx.x*256 + threadIdx.x;
    if (idx >= NB_*NP_) return;
    int b = idx / NP_;
    float m = __uint_as_float(mx[b]);
    float v = 20.0f * log10f(amp[idx] / m);
    out[idx] = fminf(0.0f, fmaxf(-60.0f, v));
}

// ------------------------------------------------------------------
extern "C" void kernel_launch(void* const* d_in, const int* in_sizes, int n_in,
                              void* d_out, int out_size, void* d_ws, size_t ws_size,
                              hipStream_t stream) {
    const float* datas   = (const float*)d_in[0];
    // d_in[1] = locs (unused by reference too)
    const float* delays  = (const float*)d_in[2];
    const float* weights = (const float*)d_in[3];
    const float* bf      = (const float*)d_in[4];
    float* out = (float*)d_out;

    char* ws = (char*)d_ws;
    size_t off = 0;
    float2* F    = (float2*)(ws + off); off += (size_t)NS*SIGS*sizeof(float2);      // spectra; reused as Z
    float2* Y    = (float2*)(ws + off); off += (size_t)NS*(NE_*2*NR_)*sizeof(float2);
    float2* Hinv = (float2*)(ws + off); off += (size_t)NS*NT_*NE_*sizeof(float2);
    float2* IQ   = (float2*)(ws + off); off += (size_t)NS*SIGS*sizeof(float2);
    float*  amp  = (float*)(ws + off);  off += (size_t)NB_*NP_*sizeof(float);
    unsigned int* mx = (unsigned int*)(ws + off);

    hipMemsetAsync(mx, 0, NB_*sizeof(unsigned int), stream);

    k_h_hinv  <<<NS, 256, 0, stream>>>(delays, weights, Hinv);
    k_fft_fwd <<<SIGS, 256, 0, stream>>>(datas, F);
    k_encode  <<<dim3(NS, NB_), 256, 0, stream>>>(delays, weights, F, Y);
    k_herm    <<<((NS/2 + 1)*4096 + 255)/256, 256, 0, stream>>>(Y);
    k_decode  <<<dim3(NS, NB_), 256, 0, stream>>>(Hinv, Y, F);   // Z overwrites F
    k_hermhilb<<<((NS/2 + 1)*SIGS + 255)/256, 256, 0, stream>>>(F);
    k_ifft    <<<SIGS, 256, 0, stream>>>(F, IQ);
    k_beamform<<<(NB_*NP_ + 255)/256, 256, 0, stream>>>(IQ, bf, amp, mx);
    k_finalize<<<(NB_*NP_ + 255)/256, 256, 0, stream>>>(amp, mx, out);
}